// TensorExpansion_23424751632770
// MI455X (gfx1250) — compile-verified
//
#include <hip/hip_runtime.h>
#include <complex>
#include <vector>
#include <algorithm>
#include <numeric>
#include <cmath>
#include <cstring>

#define NAO 14
#define NOUT 196
#define NKSTEP 49        // 196 / 4 K-steps (exact)
#define NNT 13           // ceil(196/16) N tiles
#define MPAIRS 15625     // 500000 / 32 : two 16-row M-tiles per wave (exact)

typedef float v2f __attribute__((ext_vector_type(2)));
typedef float v8f __attribute__((ext_vector_type(8)));

// Packed B-operand: [kstep][ntile][lane][2] floats, fragment-ordered so each
// lane issues one coalesced 8B load per WMMA. 163 KB -> WGP$/L2 resident.
__device__ float g_Wpack[NKSTEP * NNT * 32 * 2];

__global__ __launch_bounds__(32) void te_wmma_kernel(
    const float* __restrict__ X, float* __restrict__ Y)
{
  const int lane  = threadIdx.x & 31;
  const int m0    = blockIdx.x * 32;    // rows m0..m0+15 (tile 0), m0+16..m0+31 (tile 1)
  const int lrow  = lane & 15;
  const int khalf = (lane >> 4) << 1;   // lanes 16..31 hold K+2/K+3 (ISA A layout)

  const float* xrow0 = X + (m0 + lrow) * NOUT + khalf;
  const float* xrow1 = xrow0 + 16 * NOUT;

  v8f acc0[NNT], acc1[NNT];
#pragma unroll
  for (int t = 0; t < NNT; ++t) {
    v8f z = {0.f, 0.f, 0.f, 0.f, 0.f, 0.f, 0.f, 0.f};
    acc0[t] = z;
    acc1[t] = z;
  }

#pragma unroll 1
  for (int ks = 0; ks < NKSTEP; ++ks) {
    // Streaming A fragments: non-temporal, read exactly once per kernel.
    v2f a0 = __builtin_nontemporal_load((const v2f*)(xrow0 + ks * 4));
    v2f a1 = __builtin_nontemporal_load((const v2f*)(xrow1 + ks * 4));
    const float* wp = g_Wpack + (ks * NNT * 32 + lane) * 2;
#pragma unroll
    for (int nt = 0; nt < NNT; ++nt) {
      v2f b = *(const v2f*)(wp + nt * 64);   // reused operand: default (RT) caching
      // D(16x16,f32) = A(16x4,f32) x B(4x16,f32) + C ; one B feeds two M-tiles
      acc0[nt] = __builtin_amdgcn_wmma_f32_16x16x4_f32(
          false, a0, false, b, (short)0, acc0[nt], false, false);
      acc1[nt] = __builtin_amdgcn_wmma_f32_16x16x4_f32(
          false, a1, false, b, (short)0, acc1[nt], false, false);
    }
  }

  // D layout: VGPR r -> lanes 0-15: (M=r, N=lane); lanes 16-31: (M=r+8, N=lane-16)
  const int mbase = m0 + ((lane >> 4) << 3);
#pragma unroll
  for (int nt = 0; nt < NNT; ++nt) {
    const int n = nt * 16 + lrow;
    if (n < NOUT) {                 // only last N-tile is partially padded
      float* y0 = Y + mbase * NOUT + n;
      float* y1 = y0 + 16 * NOUT;
#pragma unroll
      for (int r = 0; r < 8; ++r) {
        __builtin_nontemporal_store(acc0[nt][r], y0 + r * NOUT);
        __builtin_nontemporal_store(acc1[nt][r], y1 + r * NOUT);
      }
    }
  }
}

// ---------------------------------------------------------------------------
// Host-side construction of W (Wigner-3j change-of-basis), executed
// deterministically on every kernel_launch call.
// ---------------------------------------------------------------------------
namespace teconst {

typedef std::complex<double> cd;

static double factd(int n) { double r = 1.0; for (int i = 2; i <= n; ++i) r *= (double)i; return r; }

static double su2_cg(int j1, int m1, int j2, int m2, int j3, int m3) {
  if (m3 != m1 + m2) return 0.0;
  int vmin = std::max(std::max(-j1 + j2 + m3, -j1 + m1), 0);
  int vmax = std::min(std::min(j2 + j3 + m1, j3 - j1 + j2), j3 + m3);
  double c = std::sqrt((2.0 * j3 + 1.0) * factd(j3 + j1 - j2) * factd(j3 - j1 + j2) *
                       factd(j1 + j2 - j3) * factd(j3 + m3) * factd(j3 - m3) /
                       (factd(j1 + j2 + j3 + 1) * factd(j1 - m1) * factd(j1 + m1) *
                        factd(j2 - m2) * factd(j2 + m2)));
  double s = 0.0;
  for (int v = vmin; v <= vmax; ++v) {
    int e = v + j2 + m2;
    double sgn = (((e % 2) + 2) % 2) ? -1.0 : 1.0;
    s += sgn * factd(j2 + j3 + m1 - v) * factd(j1 - m1 + v) /
         (factd(v) * factd(j3 - j1 + j2 - v) * factd(j3 + m3 - v) * factd(v + j1 - j2 - m3));
  }
  return c * s;
}

static void q_r2c(int l, cd q[9][9]) {
  for (int i = 0; i < 9; ++i) for (int j = 0; j < 9; ++j) q[i][j] = cd(0.0, 0.0);
  const double s = 1.0 / std::sqrt(2.0);
  for (int m = -l; m < 0; ++m) {
    q[l + m][l - m] = cd(s, 0.0);     // col l+|m|
    q[l + m][l + m] = cd(0.0, -s);    // col l-|m|
  }
  q[l][l] = cd(1.0, 0.0);
  for (int m = 1; m <= l; ++m) {
    double sgn = (m % 2) ? -1.0 : 1.0;
    q[l + m][l + m] = cd(sgn * s, 0.0);
    q[l + m][l - m] = cd(0.0, sgn * s);
  }
  cd f(1.0, 0.0); const cd mi(0.0, -1.0);
  for (int t = 0; t < l; ++t) f *= mi;  // (-i)^l
  for (int i = 0; i < 2 * l + 1; ++i)
    for (int j = 0; j < 2 * l + 1; ++j) q[i][j] *= f;
}

static void wigner3j(int l1, int l2, int l3, double out[9][5][5]) {
  int n1 = 2 * l1 + 1, n2 = 2 * l2 + 1, n3 = 2 * l3 + 1;
  double csu2[9][5][5];
  std::memset(csu2, 0, sizeof(csu2));
  for (int m1 = -l1; m1 <= l1; ++m1)
    for (int m2 = -l2; m2 <= l2; ++m2) {
      int m3 = m1 + m2;
      if (m3 >= -l3 && m3 <= l3)
        csu2[l1 + m1][l2 + m2][l3 + m3] = su2_cg(l1, m1, l2, m2, l3, m3);
    }
  cd q1[9][9], q2[9][9], q3[9][9];
  q_r2c(l1, q1); q_r2c(l2, q2); q_r2c(l3, q3);
  cd cc[9][5][5];
  double nre = 0.0, nim = 0.0;
  for (int a = 0; a < n1; ++a)
    for (int b = 0; b < n2; ++b)
      for (int c = 0; c < n3; ++c) {
        cd s(0.0, 0.0);
        for (int i = 0; i < n1; ++i)
          for (int k = 0; k < n2; ++k)
            for (int n = 0; n < n3; ++n)
              s += q1[i][a] * q2[k][b] * std::conj(q3[n][c]) * csu2[i][k][n];
        cc[a][b][c] = s;
        nre += s.real() * s.real();
        nim += s.imag() * s.imag();
      }
  bool useRe = std::sqrt(nre) >= std::sqrt(nim);
  double nrm = std::sqrt(useRe ? nre : nim);
  std::memset(out, 0, sizeof(double) * 9 * 5 * 5);
  for (int a = 0; a < n1; ++a)
    for (int b = 0; b < n2; ++b)
      for (int c = 0; c < n3; ++c)
        out[a][b][c] = (useRe ? cc[a][b][c].real() : cc[a][b][c].imag()) / nrm;
}

struct Blk { int rs, nr, cs, nc, L, par, K; double cg[9][5][5]; };

static void build_wpack(float* hW) {
  const int LSa[6] = {0, 0, 0, 1, 1, 2};
  const int IDX_CHANGE[14] = {0, 1, 2, 5, 3, 4, 8, 6, 7, 11, 13, 9, 12, 10};
  int idx_inv[14];
  for (int t = 0; t < 14; ++t) idx_inv[IDX_CHANGE[t]] = t;

  std::vector<Blk> blks;
  blks.reserve(64);
  int rs = 0;
  for (int ii = 0; ii < 6; ++ii) {
    int li = LSa[ii], nr = 2 * li + 1;
    int cs = 0;
    for (int jj = 0; jj < 6; ++jj) {
      int lj = LSa[jj], nc = 2 * lj + 1;
      int lo = (li > lj) ? (li - lj) : (lj - li);
      for (int L = lo; L <= li + lj; ++L) {
        Blk b; b.rs = rs; b.nr = nr; b.cs = cs; b.nc = nc; b.L = L;
        b.par = ((li + lj) % 2) ? -1 : 1; b.K = 2 * L + 1;
        wigner3j(L, li, lj, b.cg);
        blks.push_back(b);
      }
      cs += nc;
    }
    rs += nr;
  }
  const int nb = (int)blks.size();
  std::vector<int> order(nb);
  std::iota(order.begin(), order.end(), 0);
  std::stable_sort(order.begin(), order.end(), [&](int a, int b) {
    if (blks[a].L != blks[b].L) return blks[a].L < blks[b].L;
    return blks[a].par < blks[b].par;
  });
  std::vector<int> colstart(nb);
  int off = 0;
  for (int t = 0; t < nb; ++t) { colstart[order[t]] = off; off += blks[order[t]].K; }
  // off == 196 by construction

  static double Wd[NOUT][NOUT];
  std::memset(Wd, 0, sizeof(Wd));
  for (int bi = 0; bi < nb; ++bi) {
    const Blk& b = blks[bi];
    for (int k = 0; k < b.K; ++k)
      for (int i = 0; i < b.nr; ++i)
        for (int j = 0; j < b.nc; ++j) {
          int oc = colstart[bi] + k;
          int ic = idx_inv[b.rs + i] * NAO + idx_inv[b.cs + j];
          Wd[oc][ic] += b.cg[k][i][j];
        }
  }

  // Pack W^T into the B-fragment layout assumed by the kernel:
  // VGPR0: lanes 0-15 -> K=4ks+0, lanes 16-31 -> K=4ks+2; VGPR1: +1. N = nt*16 + (lane&15).
  for (int ks = 0; ks < NKSTEP; ++ks)
    for (int nt = 0; nt < NNT; ++nt)
      for (int lane = 0; lane < 32; ++lane)
        for (int r = 0; r < 2; ++r) {
          int n = nt * 16 + (lane & 15);
          int k = ks * 4 + ((lane >> 4) << 1) + r;
          float v = (n < NOUT) ? (float)Wd[n][k] : 0.0f;
          hW[((ks * NNT + nt) * 32 + lane) * 2 + r] = v;
        }
}

} // namespace teconst

extern "C" void kernel_launch(void* const* d_in, const int* in_sizes, int n_in,
                              void* d_out, int out_size, void* d_ws, size_t ws_size,
                              hipStream_t stream) {
  (void)in_sizes; (void)n_in; (void)out_size; (void)d_ws; (void)ws_size;

  static float hW[NKSTEP * NNT * 32 * 2];
  teconst::build_wpack(hW);   // deterministic, recomputed every call

  void* dW = nullptr;
  hipGetSymbolAddress(&dW, HIP_SYMBOL(g_Wpack));
  hipMemcpyAsync(dW, hW, sizeof(hW), hipMemcpyHostToDevice, stream);

  const float* X = (const float*)d_in[0];
  float* Y = (float*)d_out;
  te_wmma_kernel<<<MPAIRS, 32, 0, stream>>>(X, Y);
}